// GraphNeuralNetwork_3152505996095
// MI455X (gfx1250) — compile-verified
//
#include <hip/hip_runtime.h>
#include <hip/hip_bf16.h>

// ---------------------------------------------------------------------------
// GCN forward: 3 x (GEMM -> SpMM scatter-add -> bias[+ReLU])
//   N = 100000 nodes, E = 1.6M edges, dims 128 -> 256 -> 256 -> 128, fp32.
// GEMMs use V_WMMA_F32_16X16X4_F32 (exact fp32 semantics, CDNA5 matrix pipe)
// with the W slice staged in LDS (bank-conflict-free, shared by all 8 waves).
// SpMM is edge-parallel with global f32 atomics (L2-resident: 102MB < 192MB).
// ---------------------------------------------------------------------------

typedef float v2f __attribute__((ext_vector_type(2)));
typedef float v8f __attribute__((ext_vector_type(8)));

#define GCN_N_NODES 100000
#define GCN_N_EDGES 1600000
#define GCN_IN  128
#define GCN_HID 256
#define GCN_OUT 128

// ---------------------------------------------------------------------------
// WMMA f32 GEMM:  C[M x OUT] = A[M x K] @ W[K x OUT]
//
// Block = 256 threads = 8 waves. blockIdx.y selects a 64-column n-group shared
// by the whole block; wave w owns M-tile (blockIdx.x*8 + w), a 16x64 output
// tile held in 4 f32 accumulators.
//
// W[k0:k0+128, n0:n0+64] is staged in LDS with row stride 72 floats:
// a B-fragment ds_load reads row kk on lanes 0-15 and row kk+2 on lanes 16-31;
// 2*72 = 144 = 16 (mod 64 banks) -> the half-waves hit disjoint bank ranges.
//
// Fragment layouts (wave32, ISA 7.12.2):
//   A 16x4 f32 : lanes 0-15 {M=lane, K=k+0/1}, lanes 16-31 {M=lane-16, K=k+2/3}
//   B 4x16 f32 : VGPR0 rows {k,k+2}, VGPR1 rows {k+1,k+3}, N striped 0-15
//   C 16x16 f32: VGPR r -> row r (lanes 0-15) / row r+8 (lanes 16-31)
// ---------------------------------------------------------------------------
template <int K, int OUT>
__global__ __launch_bounds__(256) void gcn_gemm_wmma_f32(
    const float* __restrict__ A, const float* __restrict__ W,
    float* __restrict__ C, int M) {
  constexpr int KC = 128;          // K-chunk staged per barrier round
  constexpr int LS = 72;           // LDS row stride (64 + 8 pad, bank-safe)
  __shared__ float ldsW[KC * LS];  // 36.9 KB

  const int wave = threadIdx.x >> 5;
  const int lane = threadIdx.x & 31;
  const int ml = lane & 15;          // M (A) / N (B,C) index within half-wave
  const int kh = (lane >> 4) << 1;   // K sub-offset: 0 (lanes 0-15) / 2 (16-31)

  const int mTile = blockIdx.x * 8 + wave;
  const int n0 = blockIdx.y * 64;
  const bool valid = mTile * 16 < M;
  const int m0 = valid ? mTile * 16 : 0;   // invalid waves compute on row 0, skip store

  const float* __restrict__ Arow = A + (long long)(m0 + ml) * K;

  v8f acc0 = {}, acc1 = {}, acc2 = {}, acc3 = {};

  for (int k0 = 0; k0 < K; k0 += KC) {
    __syncthreads();   // previous chunk's readers are done
    // Cooperative stage: 128 rows x 64 cols, float4 per thread per trip.
#pragma unroll
    for (int idx = threadIdx.x; idx < KC * 16; idx += 256) {
      const int row = idx >> 4;
      const int c4 = (idx & 15) << 2;
      float4 v = *(const float4*)(W + (long long)(k0 + row) * OUT + n0 + c4);
      *(float4*)(ldsW + row * LS + c4) = v;
    }
    __syncthreads();

#pragma unroll 4
    for (int kc = 0; kc < KC; kc += 4) {
      const int kk = k0 + kc;
      if (kk + 64 < K) {
        __builtin_prefetch(Arow + kk + 64, 0, 1);   // global_prefetch on A stream
      }
      v2f a = *(const v2f*)(Arow + kk + kh);        // global_load_b64

      const float* __restrict__ l0 = ldsW + (kc + kh) * LS + ml;  // rows kk / kk+2
      const float* __restrict__ l1 = l0 + LS;                     // rows kk+1 / kk+3
      v2f b0 = { l0[0],  l1[0]  };
      v2f b1 = { l0[16], l1[16] };
      v2f b2 = { l0[32], l1[32] };
      v2f b3 = { l0[48], l1[48] };

      acc0 = __builtin_amdgcn_wmma_f32_16x16x4_f32(false, a, false, b0, (short)0, acc0, false, false);
      acc1 = __builtin_amdgcn_wmma_f32_16x16x4_f32(false, a, false, b1, (short)0, acc1, false, false);
      acc2 = __builtin_amdgcn_wmma_f32_16x16x4_f32(false, a, false, b2, (short)0, acc2, false, false);
      acc3 = __builtin_amdgcn_wmma_f32_16x16x4_f32(false, a, false, b3, (short)0, acc3, false, false);
    }
  }

  if (!valid) return;

  // Store C tile: element (g, r) -> row m0 + r + (lane>=16)*8, col n0 + g*16 + ml.
  const int rbase = (lane >> 4) << 3;
  float* __restrict__ Crow = C + (long long)m0 * OUT + n0 + ml;
#pragma unroll
  for (int r = 0; r < 8; ++r) {
    float* p = Crow + (long long)(r + rbase) * OUT;
    p[0]  = acc0[r];
    p[16] = acc1[r];
    p[32] = acc2[r];
    p[48] = acc3[r];
  }
}

// ---------------------------------------------------------------------------
// out[i, :] = bias[:] broadcast (folds the "+ b" of each layer for free).
// ---------------------------------------------------------------------------
__global__ void gcn_bias_init(float* __restrict__ out, const float* __restrict__ bias,
                              int dimDiv4, long long total4) {
  long long t = (long long)blockIdx.x * blockDim.x + threadIdx.x;
  if (t >= total4) return;
  const int c4 = (int)(t & (dimDiv4 - 1)) << 2;
  float4 bv = *(const float4*)(bias + c4);
  *(float4*)(out + (t << 2)) = bv;
}

// ---------------------------------------------------------------------------
// Edge-parallel SpMM scatter:  out[dst[e], :] += sup[src[e], :] * w[e]
// One thread per (edge, 4-float chunk).  SHIFT = log2(dim/4).
// ---------------------------------------------------------------------------
template <int SHIFT>
__global__ void gcn_spmm_edge(const float* __restrict__ sup,
                              const int* __restrict__ src,
                              const int* __restrict__ dst,
                              const float* __restrict__ ew,
                              float* __restrict__ out, long long totalThreads) {
  long long t = (long long)blockIdx.x * blockDim.x + threadIdx.x;
  if (t >= totalThreads) return;
  const int e = (int)(t >> SHIFT);
  const int c = (int)(t & ((1 << SHIFT) - 1)) << 2;
  constexpr int DIM = 4 << SHIFT;

  const float w = ew[e];
  const long long so = (long long)src[e] * DIM + c;
  const long long do_ = (long long)dst[e] * DIM + c;

  float4 v = *(const float4*)(sup + so);
  float* p = out + do_;
  unsafeAtomicAdd(p + 0, v.x * w);   // -> global_atomic_add_f32
  unsafeAtomicAdd(p + 1, v.y * w);
  unsafeAtomicAdd(p + 2, v.z * w);
  unsafeAtomicAdd(p + 3, v.w * w);
}

// ---------------------------------------------------------------------------
// In-place ReLU over float4s.
// ---------------------------------------------------------------------------
__global__ void gcn_relu_inplace(float* __restrict__ buf, long long total4) {
  long long t = (long long)blockIdx.x * blockDim.x + threadIdx.x;
  if (t >= total4) return;
  float4* p = (float4*)(buf + (t << 2));
  float4 v = *p;
  v.x = fmaxf(v.x, 0.0f);
  v.y = fmaxf(v.y, 0.0f);
  v.z = fmaxf(v.z, 0.0f);
  v.w = fmaxf(v.w, 0.0f);
  *p = v;
}

extern "C" void kernel_launch(void* const* d_in, const int* in_sizes, int n_in,
                              void* d_out, int out_size, void* d_ws, size_t ws_size,
                              hipStream_t stream) {
  (void)in_sizes; (void)n_in; (void)out_size; (void)ws_size;

  const float* x   = (const float*)d_in[0];
  const int*   src = (const int*)d_in[1];
  const int*   dst = (const int*)d_in[2];
  const float* ew  = (const float*)d_in[3];
  const float* W1  = (const float*)d_in[4];
  const float* b1  = (const float*)d_in[5];
  const float* W2  = (const float*)d_in[6];
  const float* b2  = (const float*)d_in[7];
  const float* W3  = (const float*)d_in[8];
  const float* b3  = (const float*)d_in[9];
  float* out = (float*)d_out;

  float* buf0 = (float*)d_ws;                              // N x 256 support
  float* buf1 = buf0 + (size_t)GCN_N_NODES * GCN_HID;      // N x 256 aggregate

  const long long tot4_hid = (long long)GCN_N_NODES * GCN_HID / 4;   // 6.4M
  const long long tot4_out = (long long)GCN_N_NODES * GCN_OUT / 4;   // 3.2M
  const long long spmm_hid = (long long)GCN_N_EDGES * (GCN_HID / 4); // 102.4M
  const long long spmm_out = (long long)GCN_N_EDGES * (GCN_OUT / 4); // 51.2M

  const int TB = 256;
  auto blocks = [](long long n, int tb) { return (unsigned)((n + tb - 1) / tb); };

  // GEMM grids: 8 M-tiles per block (x), one 64-col n-group per block (y).
  const unsigned gemmX = (unsigned)((GCN_N_NODES / 16 + 7) / 8);  // 782
  dim3 gridHid(gemmX, GCN_HID / 64);   // 782 x 4
  dim3 gridOut(gemmX, GCN_OUT / 64);   // 782 x 2

  // ---- Layer 1: s1 = x @ W1 ; a1 = b1 + SpMM(s1) ; h1 = relu(a1) ----
  gcn_gemm_wmma_f32<GCN_IN, GCN_HID><<<gridHid, TB, 0, stream>>>(
      x, W1, buf0, GCN_N_NODES);
  gcn_bias_init<<<blocks(tot4_hid, TB), TB, 0, stream>>>(buf1, b1, GCN_HID / 4, tot4_hid);
  gcn_spmm_edge<6><<<blocks(spmm_hid, TB), TB, 0, stream>>>(
      buf0, src, dst, ew, buf1, spmm_hid);
  gcn_relu_inplace<<<blocks(tot4_hid, TB), TB, 0, stream>>>(buf1, tot4_hid);

  // ---- Layer 2: s2 = h1 @ W2 ; a2 = b2 + SpMM(s2) ; h2 = relu(a2) ----
  gcn_gemm_wmma_f32<GCN_HID, GCN_HID><<<gridHid, TB, 0, stream>>>(
      buf1, W2, buf0, GCN_N_NODES);
  gcn_bias_init<<<blocks(tot4_hid, TB), TB, 0, stream>>>(buf1, b2, GCN_HID / 4, tot4_hid);
  // buf1 was GEMM input; stream order makes the overwrite safe.
  gcn_spmm_edge<6><<<blocks(spmm_hid, TB), TB, 0, stream>>>(
      buf0, src, dst, ew, buf1, spmm_hid);
  gcn_relu_inplace<<<blocks(tot4_hid, TB), TB, 0, stream>>>(buf1, tot4_hid);

  // ---- Layer 3: s3 = h2 @ W3 ; out = b3 + SpMM(s3) (no activation) ----
  gcn_gemm_wmma_f32<GCN_HID, GCN_OUT><<<gridOut, TB, 0, stream>>>(
      buf1, W3, buf0, GCN_N_NODES);
  gcn_bias_init<<<blocks(tot4_out, TB), TB, 0, stream>>>(out, b3, GCN_OUT / 4, tot4_out);
  gcn_spmm_edge<5><<<blocks(spmm_out, TB), TB, 0, stream>>>(
      buf0, src, dst, ew, out, spmm_out);
}